// DbrxMoE_83330955478064
// MI455X (gfx1250) — compile-verified
//
#include <hip/hip_runtime.h>
#include <hip/hip_bf16.h>

// ---------------- problem constants (match reference) ----------------
#define HID   1024
#define FFNF  2048
#define NE    8
#define TOPK  2
#define TTOK  2048            // B*S
#define CAP   (TTOK*TOPK)     // worst case: every slot routed to one expert
#define MTILE 16              // tokens per block
#define FCHUNK 512            // F columns processed per chunk (per block)
#define XS (HID + 4)          // LDS stride for X tile (bank-conflict pad)
#define HS (FCHUNK + 4)       // LDS stride for h chunk

typedef __attribute__((ext_vector_type(2))) float v2f;
typedef __attribute__((ext_vector_type(8))) float v8f;

// fast silu: u * rcp(1+exp(-u)) * g  -- denominator in (1,inf), v_rcp_f32 is safe
__device__ __forceinline__ float silu_mul(float u, float g) {
    float d = 1.f + __expf(-u);
    return u * __builtin_amdgcn_rcpf(d) * g;
}

// ---------------- kernel 0: zero output + expert counters ----------------
__global__ void moe_zero(float* __restrict__ out, int* __restrict__ cnt) {
    if (blockIdx.x == 0 && threadIdx.x < NE) cnt[threadIdx.x] = 0;
    const size_t n4 = (size_t)TTOK * HID / 4;
    float4 z = make_float4(0.f, 0.f, 0.f, 0.f);
    for (size_t i = blockIdx.x * (size_t)blockDim.x + threadIdx.x; i < n4;
         i += (size_t)gridDim.x * blockDim.x)
        ((float4*)out)[i] = z;
}

// ---------------- kernel 1: build per-expert token lists ----------------
__global__ void moe_route(const long long* __restrict__ top_experts,
                          const float* __restrict__ top_weights,
                          int* __restrict__ cnt,
                          int* __restrict__ tok_id,
                          float* __restrict__ tok_w) {
    int t = blockIdx.x * blockDim.x + threadIdx.x;
    if (t >= TTOK) return;
    #pragma unroll
    for (int k = 0; k < TOPK; ++k) {
        int   e = (int)top_experts[t * TOPK + k];
        float w = top_weights[t * TOPK + k];
        int pos = atomicAdd(&cnt[e], 1);
        tok_id[e * CAP + pos] = t;
        tok_w [e * CAP + pos] = w;
    }
}

// ---------------- kernel 2: grouped GEMM + SwiGLU + down-proj ----------------
__launch_bounds__(128)
__global__ void moe_expert_gemm(const float* __restrict__ x,
                                const float* __restrict__ w_up,
                                const float* __restrict__ w_gate,
                                const float* __restrict__ w_down,
                                const int*   __restrict__ cnt,
                                const int*   __restrict__ tok_id,
                                const float* __restrict__ tok_w,
                                float*       __restrict__ out) {
    extern __shared__ float lds[];
    float* Xs  = lds;                         // [MTILE][XS]
    float* Hs  = Xs + MTILE * XS;             // [MTILE][HS]
    float* Wr  = Hs + MTILE * HS;             // [MTILE] row combine weights
    int*   Ids = (int*)(Wr + MTILE);          // [MTILE] token ids

    const int e    = blockIdx.y;
    const int n    = cnt[e];
    const int row0 = blockIdx.x * MTILE;
    if (row0 >= n) return;                    // uniform per block

    const int tid  = threadIdx.x;
    const int wave = tid >> 5;
    const int lane = tid & 31;
    const int half = lane >> 4;               // 0: lanes 0-15, 1: lanes 16-31
    const int l15  = lane & 15;
    const int koff = half * 2;                // ISA 16x4 f32 A/B layout: K += 2 in upper half
    const int mrow = l15;                     // A-matrix row = lane&15 in both halves

    if (tid < MTILE) {
        int r = row0 + tid;
        if (r < n) { Ids[tid] = tok_id[e * CAP + r]; Wr[tid] = tok_w[e * CAP + r]; }
        else       { Ids[tid] = -1;                  Wr[tid] = 0.f; }
    }
    __syncthreads();

    // stage X tile [16 x 1024] into LDS (zero-pad invalid rows)
    for (int idx = tid; idx < MTILE * (HID / 4); idx += 128) {
        int r  = idx / (HID / 4);
        int c4 = idx % (HID / 4);
        int t  = Ids[r];
        float4 v = make_float4(0.f, 0.f, 0.f, 0.f);
        if (t >= 0) v = ((const float4*)(x + (size_t)t * HID))[c4];
        *(float4*)&Xs[r * XS + c4 * 4] = v;
    }
    __syncthreads();

    const float* Wu = w_up   + (size_t)e * HID  * FFNF;
    const float* Wg = w_gate + (size_t)e * HID  * FFNF;
    const float* Wd = w_down + (size_t)e * FFNF * HID;

    // down-proj accumulators: this wave owns out columns [wave*256, wave*256+256)
    v8f acc[16];
    #pragma unroll
    for (int i = 0; i < 16; ++i) acc[i] = (v8f){0.f,0.f,0.f,0.f,0.f,0.f,0.f,0.f};

    for (int fc = 0; fc < FFNF; fc += FCHUNK) {
        // ---- Phase A: h[16][FCHUNK] for f in [fc, fc+FCHUNK); wave w -> 128 cols
        #pragma unroll 1
        for (int ft = 0; ft < 8; ++ft) {
            const int fcol = wave * 128 + ft * 16;          // col within chunk
            const int f0   = fc + fcol;
            v8f cu = (v8f){0.f,0.f,0.f,0.f,0.f,0.f,0.f,0.f};
            v8f cg = (v8f){0.f,0.f,0.f,0.f,0.f,0.f,0.f,0.f};
            const float* wuCol = Wu + f0 + l15;             // (k, f0+l15) -> k*FFNF
            const float* wgCol = Wg + f0 + l15;
            #pragma unroll 4
            for (int k0 = 0; k0 < HID; k0 += 4) {
                const int ka = k0 + koff;
                v2f a = *(const v2f*)&Xs[mrow * XS + ka];   // ds_load_b64
                v2f bu, bg;
                bu.x = wuCol[(size_t)(ka    ) * FFNF];
                bu.y = wuCol[(size_t)(ka + 1) * FFNF];
                bg.x = wgCol[(size_t)(ka    ) * FFNF];
                bg.y = wgCol[(size_t)(ka + 1) * FFNF];
                cu = __builtin_amdgcn_wmma_f32_16x16x4_f32(false, a, false, bu,
                                                           (short)0, cu, false, false);
                cg = __builtin_amdgcn_wmma_f32_16x16x4_f32(false, a, false, bg,
                                                           (short)0, cg, false, false);
            }
            // SwiGLU: h = silu(up) * gate ; scatter C-tile into LDS h buffer
            #pragma unroll
            for (int r = 0; r < 8; ++r) {
                int m = r + half * 8;                       // C/D row layout
                Hs[m * HS + fcol + l15] = silu_mul(cu[r], cg[r]);
            }
        }
        __syncthreads();

        // ---- Phase B: acc += h_chunk @ Wd[fc:fc+FCHUNK, wave's 256 cols]
        #pragma unroll
        for (int nt = 0; nt < 16; ++nt) {
            const int n0 = wave * 256 + nt * 16;
            const float* wdCol = Wd + n0 + l15;
            v8f c = acc[nt];
            #pragma unroll 4
            for (int k0 = 0; k0 < FCHUNK; k0 += 4) {
                const int ka = k0 + koff;
                v2f a = *(const v2f*)&Hs[mrow * HS + ka];   // ds_load_b64
                v2f b;
                b.x = wdCol[(size_t)(fc + ka    ) * HID];
                b.y = wdCol[(size_t)(fc + ka + 1) * HID];
                c = __builtin_amdgcn_wmma_f32_16x16x4_f32(false, a, false, b,
                                                          (short)0, c, false, false);
            }
            acc[nt] = c;
        }
        __syncthreads();                                    // before Hs is overwritten
    }

    // scale rows by combine weight, accumulate into out (tokens may hit 2 experts)
    #pragma unroll
    for (int nt = 0; nt < 16; ++nt) {
        const int col = wave * 256 + nt * 16 + l15;
        v8f c = acc[nt];
        #pragma unroll
        for (int r = 0; r < 8; ++r) {
            int m = r + half * 8;
            int t = Ids[m];
            if (t >= 0) atomicAdd(&out[(size_t)t * HID + col], c[r] * Wr[m]);
        }
    }
}

// ---------------- launcher ----------------
extern "C" void kernel_launch(void* const* d_in, const int* in_sizes, int n_in,
                              void* d_out, int out_size, void* d_ws, size_t ws_size,
                              hipStream_t stream) {
    const float*     x      = (const float*)d_in[0];
    // d_in[1] = weights (unused by reference forward)
    const float*     top_w  = (const float*)d_in[2];
    const long long* top_e  = (const long long*)d_in[3];
    const float*     w_up   = (const float*)d_in[4];
    const float*     w_gate = (const float*)d_in[5];
    const float*     w_down = (const float*)d_in[6];
    float*           out    = (float*)d_out;

    // workspace layout
    char*  ws     = (char*)d_ws;
    int*   cnt    = (int*)ws;                                   // NE ints (padded)
    int*   tok_id = (int*)(ws + 256);                           // NE*CAP ints
    float* tok_w  = (float*)(ws + 256 + (size_t)NE * CAP * sizeof(int));

    moe_zero <<<512, 256, 0, stream>>>(out, cnt);
    moe_route<<<(TTOK + 255) / 256, 256, 0, stream>>>(top_e, top_w, cnt, tok_id, tok_w);

    size_t shmem = (size_t)(MTILE * XS + MTILE * HS + MTILE) * sizeof(float)
                 + (size_t)MTILE * sizeof(int);
    moe_expert_gemm<<<dim3(CAP / MTILE, NE, 1), 128, shmem, stream>>>(
        x, w_up, w_gate, w_down, cnt, tok_id, tok_w, out);
}